// FPSampling_84052509982732
// MI455X (gfx1250) — compile-verified
//
#include <hip/hip_runtime.h>

// Farthest Point Sampling for MI455X (gfx1250).
//
// Mapping: one 1024-thread workgroup (32 wave32 waves = 8 waves/SIMD) per
// batch, persistent across all m-1 strictly sequential FPS steps.
//  - The full per-batch distance field (n floats = 256KB for n=65536) is
//    resident in LDS -- possible only because CDNA5 has 320KB LDS per WGP.
//  - Points (768KB/batch) stream from the 192MB L2 every step (6MB total
//    working set; HBM is touched once).
//  - Each thread owns slots i = tid mod 1024: the scan is race-free and the
//    three q[0..2] loads merge into one global_load_b96 per point.
//  - Block argmax: pack (f32 bits << 32) | ~idx (distances >= 0, so IEEE bit
//    order == unsigned order; ~idx makes ties prefer the smallest index,
//    matching jnp.argmax), wave32 __shfl_xor reduction -> LDS -> wave-0
//    reduction, split workgroup barriers between phases.
//  - The selected pivot index is made wave-uniform via readfirstlane so the
//    per-step pivot fetch can use the scalar path.

#define BLOCK 1024
#define NWAVES (BLOCK / 32)

__device__ __forceinline__ unsigned long long shflxor_u64(unsigned long long v, int m) {
  unsigned lo = (unsigned)__shfl_xor((int)(unsigned)(v & 0xffffffffull), m, 32);
  unsigned hi = (unsigned)__shfl_xor((int)(unsigned)(v >> 32), m, 32);
  return ((unsigned long long)hi << 32) | lo;
}

// Block-wide argmax from per-thread (bestV, bestI); thread 0 emits outputs.
// Returns the winning local index (wave-uniform).
__device__ __forceinline__ int block_argmax_and_emit(
    float bestV, unsigned bestI, unsigned long long* red, unsigned int* bcast,
    const float* __restrict__ pb, long long gbase,
    float* __restrict__ np_out, float* __restrict__ idx_out,
    int b, int m, int s, int tid) {
  unsigned long long key =
      ((unsigned long long)__float_as_uint(bestV) << 32) | (unsigned)(~bestI);
#pragma unroll
  for (int off = 16; off > 0; off >>= 1) {
    unsigned long long v = shflxor_u64(key, off);
    key = (v > key) ? v : key;
  }
  if ((tid & 31) == 0) red[tid >> 5] = key;
  __syncthreads();

  if (tid < NWAVES) {
    unsigned long long k2 = red[tid];
#pragma unroll
    for (int off = 16; off > 0; off >>= 1) {
      unsigned long long v = shflxor_u64(k2, off);
      k2 = (v > k2) ? v : k2;
    }
    if (tid == 0) {
      unsigned j = ~(unsigned)(k2 & 0xffffffffull);
      *bcast = j;
      idx_out[(long long)b * m + s] = (float)(gbase + (long long)j);
      const float* __restrict__ qj = pb + 3 * (long long)j;
      long long r = ((long long)b * m + s) * 3;
      np_out[r + 0] = qj[0];
      np_out[r + 1] = qj[1];
      np_out[r + 2] = qj[2];
    }
  }
  __syncthreads();
  return __builtin_amdgcn_readfirstlane((int)*bcast);  // uniform pivot
}

__device__ __forceinline__ void emit_first_sample(
    const float* __restrict__ pb, long long gbase,
    float* __restrict__ np_out, float* __restrict__ no_out,
    float* __restrict__ idx_out, int b, int m) {
  no_out[b] = (float)((long long)(b + 1) * m);
  idx_out[(long long)b * m] = (float)gbase;  // first sample: local idx 0
  long long r0 = (long long)b * m * 3;
  np_out[r0 + 0] = pb[0];
  np_out[r0 + 1] = pb[1];
  np_out[r0 + 2] = pb[2];
}

__global__ void __launch_bounds__(BLOCK, 1)
fps_kernel(const float* __restrict__ p, const int* __restrict__ o,
           float* __restrict__ out, int n, int m, int B) {
  extern __shared__ float dists[];            // n floats (256KB for n=65536)
  __shared__ unsigned long long red[NWAVES];  // per-wave argmax keys
  __shared__ unsigned int bcast;              // winning local index

  const int tid = threadIdx.x;
  const int b = blockIdx.x;

  const long long pbase = (long long)b * n;
  const float* __restrict__ pb = p + pbase * 3;
  const long long gbase = (b == 0) ? 0 : (long long)o[b - 1];  // idx base per ref

  float* __restrict__ np_out  = out;                          // [B*m, 3]
  float* __restrict__ no_out  = out + (long long)B * m * 3;   // [B]
  float* __restrict__ idx_out = no_out + B;                   // [B*m]

  // Init distance field to BIG (reference uses 1e10).
  for (int i = tid; i < n; i += BLOCK) dists[i] = 1.0e10f;

  if (tid == 0) emit_first_sample(pb, gbase, np_out, no_out, idx_out, b, m);

  // Warm the point stream (global_prefetch_b8 on gfx1250).
  __builtin_prefetch(pb + 3 * (long long)tid, 0, 0);

  int last = 0;
  __syncthreads();

  for (int s = 1; s < m; ++s) {
    const float* __restrict__ q0 = pb + 3 * (long long)last;  // scalarizable
    const float lx = q0[0];
    const float ly = q0[1];
    const float lz = q0[2];

    // Distance update + thread-local argmax over owned slots (i = tid mod 1024).
    float bestV = 0.0f;                 // dists >= 0, so 0 is a safe floor
    unsigned bestI = (unsigned)tid;
#pragma unroll 8
    for (int i = tid; i < n; i += BLOCK) {
      const float* __restrict__ q = pb + 3 * (long long)i;  // global_load_b96
      float dx = q[0] - lx;
      float dy = q[1] - ly;
      float dz = q[2] - lz;
      float d  = fmaf(dx, dx, fmaf(dy, dy, dz * dz));
      float nd = fminf(dists[i], d);
      dists[i] = nd;
      if (nd > bestV) { bestV = nd; bestI = (unsigned)i; }  // strict > => first idx
    }

    last = block_argmax_and_emit(bestV, bestI, red, &bcast, pb, gbase,
                                 np_out, idx_out, b, m, s, tid);
  }
}

extern "C" void kernel_launch(void* const* d_in, const int* in_sizes, int n_in,
                              void* d_out, int out_size, void* d_ws, size_t ws_size,
                              hipStream_t stream) {
  (void)n_in; (void)d_ws; (void)ws_size;
  const float* p = (const float*)d_in[0];
  const int*   o = (const int*)d_in[1];

  const int N = in_sizes[0] / 3;          // total points
  const int B = in_sizes[1];              // batches
  const int n = N / B;                    // points per batch
  // out_size = 4*m_total + B  (n_p: 3*m_total, n_o: B, idx: m_total)
  const int m_total = (out_size - B) / 4;
  const int m = m_total / B;              // samples per batch

  const size_t lds_bytes = (size_t)n * sizeof(float);  // distance field in LDS
  fps_kernel<<<dim3(B), dim3(BLOCK), lds_bytes, stream>>>(p, o, (float*)d_out,
                                                          n, m, B);
}